// CRF_51067161149668
// MI455X (gfx1250) — compile-verified
//
#include <hip/hip_runtime.h>
#include <hip/hip_bf16.h>

// CRF NLL on gfx1250: forward algorithm as shifted-exp GEMM on V_WMMA_F32_16X16X4_F32.
// B=512 sequences, T=512 steps, K=64 tags. START=62, STOP=63.
//
// Per-step latency optimizations (only 32 workgroups exist -> latency-bound):
//  * one barrier per timestep (stale-exact-max shift + double-buffered LDS)
//  * software-pipelined emission loads (HBM latency overlapped with WMMA chain)
//  * raw v_log_f32 / v_exp_f32 (no denorm-guard VALU; v_nop covers TRANS hazard)

#define KDIM 64
#define TDIM 512
#define BDIM 512
#define START_TAG 62
#define STOP_TAG 63
#define NEGV (-10000.0f)
#define LN2F 0.6931471805599453f
#define LOG2EF 1.4426950408889634f

typedef __attribute__((ext_vector_type(2))) float v2f;
typedef __attribute__((ext_vector_type(8))) float v8f;

__device__ __forceinline__ float vlog2f(float x) {
    float r;
    asm("v_log_f32 %0, %1\n\tv_nop" : "=v"(r) : "v"(x));
    return r;
}
__device__ __forceinline__ float vexp2f(float x) {
    float r;
    asm("v_exp_f32 %0, %1\n\tv_nop" : "=v"(r) : "v"(x));
    return r;
}

// ---------------------------------------------------------------------------
// Forward algorithm. One workgroup (4 waves, 128 thr) per 16 batch rows.
// Wave w owns output tags [16w, 16w+16). Eexp^T slice lives in B-matrix regs.
// p (= exp(alpha_t - rowmax(alpha_{t-1}))) exchanged via double-buffered LDS,
// column-major p[j*16 + m].
// ---------------------------------------------------------------------------
__global__ __launch_bounds__(128)
void crf_forward_kernel(const float* __restrict__ feats,
                        const float* __restrict__ transitions,
                        const int*   __restrict__ lengths,
                        float*       __restrict__ fs_out) {
    __shared__ float p_lds[2][16 * KDIM];   // double-buffered p, column-major [j][m]
    __shared__ float maxbuf[2][4][16];      // double-buffered per-wave partial row maxima
    __shared__ float term_lds[16 * KDIM];   // terminal scores, column-major

    const int tid   = threadIdx.x;
    const int wave  = tid >> 5;
    const int lane  = tid & 31;
    const int lhalf = lane >> 4;            // 0: lanes 0-15, 1: lanes 16-31
    const int l16   = lane & 15;
    const int base  = blockIdx.x * 16;      // first batch row of this tile
    const int ntag  = 16 * wave + l16;      // this lane's output tag (C/D: N = lane%16)

    // ---- B-matrix registers: B[k][n] = exp(trans[n,k]) = Eexp^T, chunked K=4.
    // B chunk c: VGPR0 holds k = 4c + 2*lhalf, VGPR1 holds k = 4c + 2*lhalf + 1.
    v2f Breg[16];
#pragma unroll
    for (int c = 0; c < 16; ++c) {
        int k0 = 4 * c + 2 * lhalf;
        Breg[c].x = __expf(transitions[ntag * KDIM + k0]);
        Breg[c].y = __expf(transitions[ntag * KDIM + k0 + 1]);
    }
    const float transStop = transitions[STOP_TAG * KDIM + ntag];

    // lengths for the rows this lane's accumulator VGPRs refer to
    int lenr[8];
#pragma unroll
    for (int v = 0; v < 8; ++v) lenr[v] = lengths[base + v + 8 * lhalf];

    // per-row emission base pointer (row m = v + 8*lhalf handled as v*T*K offset)
    const float* fbase = feats + (size_t)(base + 8 * lhalf) * TDIM * KDIM + ntag;

    // ---- init: alpha_0 = NEG except START=0  =>  rowmax=0, p1[m,j] = (j==START)
    for (int idx = tid; idx < 16 * KDIM; idx += 128) {
        int j = idx >> 4;
        p_lds[0][idx] = (j == START_TAG) ? 1.0f : 0.0f;
    }
    ((float*)maxbuf)[tid] = 0.0f;           // 128 floats = both buffers; buf0 = max(alpha_0)=0

    float rowMshift[8], fv[8], featc[8];
#pragma unroll
    for (int v = 0; v < 8; ++v) {
        rowMshift[v] = 0.0f;                // shift used inside p_lds[0]
        fv[v] = NEGV;
        featc[v] = fbase[(size_t)v * TDIM * KDIM];      // emissions for t=1
    }
    __syncthreads();

    for (int t = 1; t <= TDIM; ++t) {
        const int pr = (t - 1) & 1;         // buffers written last step
        const int pw = t & 1;               // buffers written this step

        // ---- exact row max of alpha_{t-1} (one-step-stale shift for p_{t+1})
        float Mst[8];
#pragma unroll
        for (int v = 0; v < 8; ++v) {
            int m = v + 8 * lhalf;
            Mst[v] = fmaxf(fmaxf(maxbuf[pr][0][m], maxbuf[pr][1][m]),
                           fmaxf(maxbuf[pr][2][m], maxbuf[pr][3][m]));
        }

        // ---- software-pipelined emission loads for t+1 (clamped; unused at t=T)
        float featn[8];
        {
            int tn = (t < TDIM) ? t : (TDIM - 1);
#pragma unroll
            for (int v = 0; v < 8; ++v)
                featn[v] = fbase[(size_t)v * TDIM * KDIM + (size_t)tn * KDIM];
        }

        // ---- GEMM: s[m][n] = sum_j p[m][j] * B[j][n] via 16 x WMMA_F32_16X16X4_F32
        v8f acc = {};
#pragma unroll
        for (int c = 0; c < 16; ++c) {
            // A chunk: lanes hold M = lane%16; VGPR0: k = 4c+2*lhalf, VGPR1: +1
            int addr = (4 * c + 2 * lhalf) * 16 + l16;
            v2f a;
            a.x = p_lds[pr][addr];
            a.y = p_lds[pr][addr + 16];
            acc = __builtin_amdgcn_wmma_f32_16x16x4_f32(
                false, a, false, Breg[c], (short)0, acc, false, false);
        }

        // ---- alpha_t = shift + ln(s) + emit ; snapshot fv at t==len
        float alpha[8];
#pragma unroll
        for (int v = 0; v < 8; ++v) {
            float av = fmaf(vlog2f(acc[v]), LN2F, rowMshift[v] + featc[v]);
            alpha[v] = av;
            if (t == lenr[v]) fv[v] = av;
        }

        // ---- this wave's 16-tag partial row max of alpha_t -> maxbuf[pw]
        float pmax[8];
#pragma unroll
        for (int v = 0; v < 8; ++v) {
            float mx = alpha[v];
            mx = fmaxf(mx, __shfl_xor(mx, 1, 32));
            mx = fmaxf(mx, __shfl_xor(mx, 2, 32));
            mx = fmaxf(mx, __shfl_xor(mx, 4, 32));
            mx = fmaxf(mx, __shfl_xor(mx, 8, 32));
            pmax[v] = mx;
        }
        if (l16 == 0) {
#pragma unroll
            for (int v = 0; v < 8; ++v)
                maxbuf[pw][wave][v + 8 * lhalf] = pmax[v];
        }

        // ---- p_{t+1} = exp(alpha_t - Mst) -> p_lds[pw]; roll pipeline registers
#pragma unroll
        for (int v = 0; v < 8; ++v) {
            int m = v + 8 * lhalf;
            p_lds[pw][ntag * 16 + m] = vexp2f((alpha[v] - Mst[v]) * LOG2EF);
            rowMshift[v] = Mst[v];
            featc[v] = featn[v];
        }

        __syncthreads();   // single barrier: pr-buffers fully read, pw-buffers fully written
    }

    // ---- terminal = fv + trans[STOP,:]; per-row logsumexp over all 64 tags
#pragma unroll
    for (int v = 0; v < 8; ++v) {
        int m = v + 8 * lhalf;
        term_lds[ntag * 16 + m] = fv[v] + transStop;
    }
    __syncthreads();
    if (tid < 16) {
        float mx = -__builtin_inff();
        for (int j = 0; j < KDIM; ++j) mx = fmaxf(mx, term_lds[j * 16 + tid]);
        float s = 0.0f;
        for (int j = 0; j < KDIM; ++j) s += __expf(term_lds[j * 16 + tid] - mx);
        fs_out[base + tid] = mx + __logf(s);
    }
}

// ---------------------------------------------------------------------------
// Gold-path score: one thread per sequence (tiny vs. the forward pass).
// ---------------------------------------------------------------------------
__global__ __launch_bounds__(256)
void crf_gold_kernel(const float* __restrict__ feats,
                     const float* __restrict__ transitions,
                     const int*   __restrict__ tags,
                     const int*   __restrict__ lengths,
                     float*       __restrict__ gold_out) {
    int b = blockIdx.x * blockDim.x + threadIdx.x;
    if (b >= BDIM) return;
    int len = lengths[b];
    int prev = START_TAG;
    float score = 0.0f;
    for (int t = 0; t < TDIM; ++t) {
        if (t < len) {
            int tag = tags[b * TDIM + t];
            score += transitions[tag * KDIM + prev]
                   + feats[((size_t)b * TDIM + t) * (size_t)KDIM + tag];
            prev = tag;
        }
    }
    score += transitions[STOP_TAG * KDIM + prev];
    gold_out[b] = score;
}

// ---------------------------------------------------------------------------
// mean(forward - gold): deterministic LDS tree reduction, single block.
// ---------------------------------------------------------------------------
__global__ __launch_bounds__(512)
void crf_reduce_kernel(const float* __restrict__ fs,
                       const float* __restrict__ gold,
                       float* __restrict__ out) {
    __shared__ float sm[512];
    int i = threadIdx.x;
    sm[i] = fs[i] - gold[i];
    __syncthreads();
    for (int s = 256; s > 0; s >>= 1) {
        if (i < s) sm[i] += sm[i + s];
        __syncthreads();
    }
    if (i == 0) out[0] = sm[0] * (1.0f / (float)BDIM);
}

// ---------------------------------------------------------------------------
extern "C" void kernel_launch(void* const* d_in, const int* in_sizes, int n_in,
                              void* d_out, int out_size, void* d_ws, size_t ws_size,
                              hipStream_t stream) {
    const float* feats       = (const float*)d_in[0];  // (B, T, K) f32
    const float* transitions = (const float*)d_in[1];  // (K, K)    f32
    const int*   tags        = (const int*)d_in[2];    // (B, T)    i32
    const int*   lengths     = (const int*)d_in[3];    // (B,)      i32
    float* out = (float*)d_out;                        // scalar f32

    float* fs   = (float*)d_ws;        // [B] forward scores
    float* gold = fs + BDIM;           // [B] gold scores

    crf_gold_kernel<<<(BDIM + 255) / 256, 256, 0, stream>>>(
        feats, transitions, tags, lengths, gold);
    crf_forward_kernel<<<BDIM / 16, 128, 0, stream>>>(
        feats, transitions, lengths, fs);
    crf_reduce_kernel<<<1, 512, 0, stream>>>(fs, gold, out);
}